// attention_edge_to_node_12034498363710
// MI455X (gfx1250) — compile-verified
//
#include <hip/hip_runtime.h>
#include <hip/hip_bf16.h>
#include <cstddef>

typedef float v2f __attribute__((ext_vector_type(2)));
typedef float v8f __attribute__((ext_vector_type(8)));

#define D_MODEL 256
#define DHEAD   64
#define NN      256
#define BB      4
#define BN      (BB * NN)   // 1024 node rows

// ---------------------------------------------------------------------------
// Phase 1a: K_node / V_node = x @ W^T + b, masked.
// One wave per 16x16 output tile, V_WMMA_F32_16X16X4_F32, K-loop step 4.
// A 16x4 layout: lanes 0-15 hold K=0,1 (vgpr0,1); lanes 16-31 hold K=2,3.
// B 4x16 layout: mirrored (n = lane&15).
// C/D 16x16: vgpr v holds M = v + 8*(lane>=16), N = lane&15.
// ---------------------------------------------------------------------------
__global__ __launch_bounds__(32)
void kv_gemm_kernel(const float* __restrict__ x,
                    const float* __restrict__ Wk, const float* __restrict__ bk,
                    const float* __restrict__ Wv, const float* __restrict__ bv,
                    const float* __restrict__ mask,
                    float* __restrict__ Kn, float* __restrict__ Vn)
{
    const float* W    = (blockIdx.z == 0) ? Wk : Wv;
    const float* bias = (blockIdx.z == 0) ? bk : bv;
    float*       out  = (blockIdx.z == 0) ? Kn : Vn;

    const int m0   = blockIdx.x * 16;   // node-row tile
    const int h0   = blockIdx.y * 16;   // head-dim tile
    const int lane = threadIdx.x;       // 0..31
    const int mn   = lane & 15;
    const int kk   = (lane >> 4) * 2;   // 0 or 2

    v8f acc = {};
    for (int k0 = 0; k0 < D_MODEL; k0 += 4) {
        v2f a, b;
        a.x = x[(m0 + mn) * D_MODEL + k0 + kk];
        a.y = x[(m0 + mn) * D_MODEL + k0 + kk + 1];
        // B[k][n] = W[h0+n][k0+k]   (W is (H, D) row-major)
        b.x = W[(h0 + mn) * D_MODEL + k0 + kk];
        b.y = W[(h0 + mn) * D_MODEL + k0 + kk + 1];
        acc = __builtin_amdgcn_wmma_f32_16x16x4_f32(false, a, false, b,
                                                    (short)0, acc, false, false);
    }
    const int hi = lane >> 4;
    const float bh = bias[h0 + mn];
#pragma unroll
    for (int v = 0; v < 8; ++v) {
        const int m = m0 + v + 8 * hi;
        out[m * DHEAD + h0 + mn] = (acc[v] + bh) * mask[m];
    }
}

// ---------------------------------------------------------------------------
// Phase 1b: P[bn, d] = sum_h K_node[bn, h] * Wq[h, d]   (M=1024, N=256, K=64)
// ---------------------------------------------------------------------------
__global__ __launch_bounds__(32)
void p_gemm_kernel(const float* __restrict__ Kn, const float* __restrict__ Wq,
                   float* __restrict__ P)
{
    const int m0   = blockIdx.x * 16;   // bn tile
    const int d0   = blockIdx.y * 16;   // d tile
    const int lane = threadIdx.x;
    const int mn   = lane & 15;
    const int kk   = (lane >> 4) * 2;

    v8f acc = {};
    for (int k0 = 0; k0 < DHEAD; k0 += 4) {
        v2f a, b;
        a.x = Kn[(m0 + mn) * DHEAD + k0 + kk];
        a.y = Kn[(m0 + mn) * DHEAD + k0 + kk + 1];
        // B[k][n] = Wq[k0+k][d0+n]   (Wq is (H, D) row-major)
        b.x = Wq[(k0 + kk) * D_MODEL + d0 + mn];
        b.y = Wq[(k0 + kk + 1) * D_MODEL + d0 + mn];
        acc = __builtin_amdgcn_wmma_f32_16x16x4_f32(false, a, false, b,
                                                    (short)0, acc, false, false);
    }
    const int hi = lane >> 4;
#pragma unroll
    for (int v = 0; v < 8; ++v) {
        const int m = m0 + v + 8 * hi;
        P[m * D_MODEL + d0 + mn] = acc[v];
    }
}

// ---------------------------------------------------------------------------
// Phase 1c: c[bn] = dot(bq, K_node[bn, :])
// ---------------------------------------------------------------------------
__global__ void cvec_kernel(const float* __restrict__ Kn,
                            const float* __restrict__ bq,
                            float* __restrict__ cvec)
{
    const int bn = blockIdx.x * blockDim.x + threadIdx.x;
    if (bn < BN) {
        float s = 0.f;
#pragma unroll
        for (int h = 0; h < DHEAD; ++h) s += bq[h] * Kn[bn * DHEAD + h];
        cvec[bn] = s;
    }
}

// ---------------------------------------------------------------------------
// Passthrough: out_x = x  (1 MB, float4)
// ---------------------------------------------------------------------------
__global__ void copy_x_kernel(const float4* __restrict__ src,
                              float4* __restrict__ dst, int n4)
{
    const int idx = blockIdx.x * blockDim.x + threadIdx.x;
    if (idx < n4) dst[idx] = src[idx];
}

// ---------------------------------------------------------------------------
// Phase 2 (bandwidth-bound): one block per (b,i); 8 waves, one edge j per wave.
// Each wave streams the 1 KB e-row, dots against P[b,j] (L2) and P[b,i] (regs),
// wave32 xor-reduce, 2-way softmax, writes 256 B of e_out.
// ---------------------------------------------------------------------------
__global__ __launch_bounds__(256)
void edge_attn_kernel(const float* __restrict__ e, const float* __restrict__ P,
                      const float* __restrict__ cvec, const float* __restrict__ Vn,
                      const float* __restrict__ mask, float* __restrict__ e_out)
{
    const int bn_i = blockIdx.x;          // b*256 + i
    const int b    = bn_i >> 8;
    const int lane = threadIdx.x & 31;
    const int wave = threadIdx.x >> 5;    // 0..7

    const float m_i = mask[bn_i];
    const float c_i = cvec[bn_i];

    // P[b,i,:] held in registers: 8 floats per lane (covers d = lane*8 .. +7)
    const float4* Pi4 = (const float4*)(P + (size_t)bn_i * D_MODEL + lane * 8);
    const float4 pi0 = Pi4[0], pi1 = Pi4[1];
    const float2 vi  = *(const float2*)(Vn + (size_t)bn_i * DHEAD + lane * 2);

    for (int j = wave; j < NN; j += 8) {
        const int bn_j = (b << 8) + j;
        const float* erow = e + ((size_t)bn_i * NN + j) * D_MODEL;
        if (j + 8 < NN)
            __builtin_prefetch(erow + 8 * D_MODEL + lane * 8, 0, 1);

        const float4 e0 = *(const float4*)(erow + lane * 8);
        const float4 e1 = *(const float4*)(erow + lane * 8 + 4);
        const float4* Pj4 = (const float4*)(P + (size_t)bn_j * D_MODEL + lane * 8);
        const float4 pj0 = Pj4[0], pj1 = Pj4[1];

        float si = e0.x*pj0.x + e0.y*pj0.y + e0.z*pj0.z + e0.w*pj0.w
                 + e1.x*pj1.x + e1.y*pj1.y + e1.z*pj1.z + e1.w*pj1.w;
        float sj = e0.x*pi0.x + e0.y*pi0.y + e0.z*pi0.z + e0.w*pi0.w
                 + e1.x*pi1.x + e1.y*pi1.y + e1.z*pi1.z + e1.w*pi1.w;

#pragma unroll
        for (int off = 16; off >= 1; off >>= 1) {
            si += __shfl_xor(si, off, 32);
            sj += __shfl_xor(sj, off, 32);
        }

        const float m_j = mask[bn_j];
        const float mm  = m_i * m_j;
        const bool att  = mm > 0.f;
        float ai = att ? __expf((si + cvec[bn_j]) * mm * 0.125f) : 1e-9f;
        float aj = att ? __expf((sj + c_i)       * mm * 0.125f) : 1e-9f;
        const float ssum = att ? (ai + aj) : 1e-9f;
        const float inv  = 1.0f / ssum;
        ai *= inv; aj *= inv;

        const float2 vj = *(const float2*)(Vn + (size_t)bn_j * DHEAD + lane * 2);
        float2 o;
        o.x = (ai * vj.x + aj * vi.x) * mm;
        o.y = (ai * vj.y + aj * vi.y) * mm;
        *(float2*)(e_out + ((size_t)bn_i * NN + j) * DHEAD + lane * 2) = o;
    }
}

// ---------------------------------------------------------------------------
extern "C" void kernel_launch(void* const* d_in, const int* in_sizes, int n_in,
                              void* d_out, int out_size, void* d_ws, size_t ws_size,
                              hipStream_t stream)
{
    (void)in_sizes; (void)n_in; (void)out_size; (void)ws_size;
    const float* x  = (const float*)d_in[0];
    const float* e  = (const float*)d_in[1];
    const float* nm = (const float*)d_in[2];
    const float* Wq = (const float*)d_in[3];
    const float* bq = (const float*)d_in[4];
    const float* Wk = (const float*)d_in[5];
    const float* bk = (const float*)d_in[6];
    const float* Wv = (const float*)d_in[7];
    const float* bv = (const float*)d_in[8];
    float* out = (float*)d_out;

    // Workspace layout (floats): Kn[1024*64] Vn[1024*64] P[1024*256] c[1024]
    float* ws = (float*)d_ws;
    float* Kn = ws;
    float* Vn = ws + 65536;
    float* P  = ws + 131072;
    float* cv = ws + 393216;

    kv_gemm_kernel<<<dim3(BN / 16, DHEAD / 16, 2), 32, 0, stream>>>(
        x, Wk, bk, Wv, bv, nm, Kn, Vn);
    p_gemm_kernel<<<dim3(BN / 16, D_MODEL / 16), 32, 0, stream>>>(Kn, Wq, P);
    cvec_kernel<<<BN / 256, 256, 0, stream>>>(Kn, bq, cv);

    copy_x_kernel<<<256, 256, 0, stream>>>((const float4*)x, (float4*)out, 65536);
    edge_attn_kernel<<<BN, 256, 0, stream>>>(e, P, cv, Vn, nm, out + BN * D_MODEL);
}